// FinalLayer_463856468357
// MI455X (gfx1250) — compile-verified
//
#include <hip/hip_runtime.h>

typedef __attribute__((ext_vector_type(2))) float v2f;
typedef __attribute__((ext_vector_type(8))) float v8f;
typedef __attribute__((ext_vector_type(4))) int   v4i;

#define BATCH  8
#define NPTS   512
#define MPTS   1024
#define CCH    16
#define WAVES  8              // waves per block (256 threads, wave32)
#define NCHUNK (NPTS / WAVES) // 64 n per wave

// ---- CDNA5 async global->LDS copy (ASYNCcnt path) --------------------------
#if __has_builtin(__builtin_amdgcn_global_load_async_to_lds_b128)
#define ASYNC_COPY_B128(gptr, lptr)                                             \
    __builtin_amdgcn_global_load_async_to_lds_b128(                             \
        (__attribute__((address_space(1))) v4i*)(gptr),                         \
        (__attribute__((address_space(3))) v4i*)(lptr), 0, 0)
#else
#define ASYNC_COPY_B128(gptr, lptr)                                             \
    do {                                                                        \
        uint32_t _l = (uint32_t)(uintptr_t)(__attribute__((address_space(3))) void*)(lptr); \
        uint64_t _g = (uint64_t)(uintptr_t)(gptr);                              \
        uint32_t _z = 0;                                                        \
        asm volatile("global_load_async_to_lds_b128 %0, %1, %2"                 \
                     :: "v"(_l), "v"(_z), "s"(_g) : "memory");                  \
    } while (0)
#endif

static __device__ __forceinline__ void async_wait_all() {
#if __has_builtin(__builtin_amdgcn_s_wait_asynccnt)
    __builtin_amdgcn_s_wait_asynccnt(0);
#else
    asm volatile("s_wait_asynccnt 0" ::: "memory");
#endif
    asm volatile("" ::: "memory"); // keep LDS reads below the wait
}

// out[b,m] = g_b + sum_c g_w[c] * sum_n y[b,n,c] * exp(-0.5*(x[b,n]-t[b,m])^2 * exp(-2*sigma[c]))
__global__ __launch_bounds__(256) void rbf_final_layer_kernel(
    const float* __restrict__ x,     // (B,N)
    const float* __restrict__ y,     // (B,N,C)
    const float* __restrict__ t,     // (B,M)
    const float* __restrict__ sigma, // (C)
    const float* __restrict__ g_w,   // (C)   (OUT_CHANNELS=1)
    const float* __restrict__ g_b,   // (1)
    float* __restrict__ out)         // (B,M)
{
    __shared__ __align__(16) float sh_y[WAVES * NCHUNK * CCH]; // 32 KB staged y
    __shared__ __align__(16) float red[WAVES * 32 * 8];        // 8 KB partials

    const int tid  = threadIdx.x;
    const int wave = tid >> 5;     // wave32 on gfx1250
    const int lane = tid & 31;
    const int hf   = lane >> 4;    // 16-lane half = K-pair selector in WMMA A layout
    const int mloc = lane & 15;    // row M within the 16-wide m tile

    const int tile = blockIdx.x & (MPTS / 16 - 1); // 64 tiles per batch
    const int b    = blockIdx.x >> 6;

    const float* xb = x + b * NPTS;
    const float* yb = y + (size_t)b * NPTS * CCH;
    const int n0 = wave * NCHUNK;

    // ---- stage this wave's 64x16 y-chunk into LDS via async b128 copies ----
    {
        const float* gchunk = yb + n0 * CCH;              // 1024 floats
        float*       lchunk = sh_y + wave * (NCHUNK * CCH);
#pragma unroll
        for (int i = 0; i < 8; ++i) {
            const int e = i * 128 + lane * 4;             // float index, 16B granules
            ASYNC_COPY_B128(gchunk + e, lchunk + e);
        }
    }

    const float tm = t[b * MPTS + tile * 16 + mloc];

    // Lane's 8 channels: c(j) = 4*(j>>1) + 2*hf + (j&1)  (native WMMA f32 16x4 A layout)
    const float LOG2E = 1.4426950408889634f;
    float beta2[8];
#pragma unroll
    for (int j = 0; j < 8; ++j) {
        int c = 4 * (j >> 1) + 2 * hf + (j & 1);
        // 0.5*log2(e) * exp(-2*sigma) ; exp(-2s) = 2^(-2s*log2e)
        beta2[j] = 0.5f * LOG2E * __builtin_amdgcn_exp2f(-2.0f * sigma[c] * LOG2E);
    }

    float acc[8];
#pragma unroll
    for (int j = 0; j < 8; ++j) acc[j] = 0.0f;

    async_wait_all(); // wave reads only its own staged chunk: no barrier needed

    const float* lchunk = sh_y + wave * (NCHUNK * CCH);
#pragma unroll 2
    for (int nn = 0; nn < NCHUNK; ++nn) {
        const float dx = xb[n0 + nn] - tm;
        const float d  = dx * dx;
        const float* lyr = lchunk + nn * CCH + 2 * hf;
#pragma unroll
        for (int p = 0; p < 4; ++p) {
            v2f yv = *(const v2f*)(lyr + 4 * p);                      // ds_load_b64
            float w0 = __builtin_amdgcn_exp2f(-beta2[2 * p]     * d); // v_exp_f32
            float w1 = __builtin_amdgcn_exp2f(-beta2[2 * p + 1] * d);
            acc[2 * p]     = fmaf(yv.x, w0, acc[2 * p]);
            acc[2 * p + 1] = fmaf(yv.y, w1, acc[2 * p + 1]);
        }
    }

    // Deterministic cross-wave reduction via LDS (fixed summation order).
    float4* dst = (float4*)&red[(wave * 32 + lane) * 8];
    dst[0] = make_float4(acc[0], acc[1], acc[2], acc[3]);
    dst[1] = make_float4(acc[4], acc[5], acc[6], acc[7]);
    __syncthreads();

    if (tid < 32) { // wave 0, EXEC all-1s for WMMA
        float tot[8];
#pragma unroll
        for (int j = 0; j < 8; ++j) tot[j] = 0.0f;
#pragma unroll
        for (int w = 0; w < WAVES; ++w) {
            const float* src = &red[(w * 32 + lane) * 8];
#pragma unroll
            for (int j = 0; j < 8; ++j) tot[j] += src[j];
        }

        // Epilogue GEMM on the matrix core: D = A(16m x 16c) * B(16c x 16) + C
        // A = per-lane accumulators (layout-native); B = g_w broadcast across all
        // 16 columns (value depends only on K); C = g_b everywhere.
        const float gb = g_b[0];
        v8f cacc;
#pragma unroll
        for (int j = 0; j < 8; ++j) cacc[j] = gb;

#pragma unroll
        for (int cc = 0; cc < 4; ++cc) {
            v2f a, bm;
            a.x  = tot[2 * cc];               // K = 4cc + 2*hf
            a.y  = tot[2 * cc + 1];           // K = 4cc + 2*hf + 1
            bm.x = g_w[4 * cc + 2 * hf];      // B row K (all N columns identical)
            bm.y = g_w[4 * cc + 2 * hf + 1];
            cacc = __builtin_amdgcn_wmma_f32_16x16x4_f32(
                /*neg_a=*/false, a, /*neg_b=*/false, bm,
                /*c_mod=*/(short)0, cacc, /*reuse_a=*/false, /*reuse_b=*/false);
        }

        // D layout: VGPR j holds (M=j, lanes 0-15) and (M=j+8, lanes 16-31); all
        // N columns equal, so lane 0 writes rows 0..7, lane 16 rows 8..15.
        float* ob = out + b * MPTS + tile * 16;
        if (lane == 0) {
#pragma unroll
            for (int j = 0; j < 8; ++j) ob[j] = cacc[j];
        } else if (lane == 16) {
#pragma unroll
            for (int j = 0; j < 8; ++j) ob[8 + j] = cacc[j];
        }
    }
}

extern "C" void kernel_launch(void* const* d_in, const int* in_sizes, int n_in,
                              void* d_out, int out_size, void* d_ws, size_t ws_size,
                              hipStream_t stream) {
    const float* x     = (const float*)d_in[0]; // (8,512,1)
    const float* y     = (const float*)d_in[1]; // (8,512,16)
    const float* t     = (const float*)d_in[2]; // (8,1024,1)
    const float* sigma = (const float*)d_in[3]; // (16)
    const float* g_w   = (const float*)d_in[4]; // (1,16)
    const float* g_b   = (const float*)d_in[5]; // (1)
    float* out = (float*)d_out;                 // (8,1024,1)

    const int blocks = BATCH * (MPTS / 16); // 512 blocks x 256 threads (8 waves)
    rbf_final_layer_kernel<<<blocks, 256, 0, stream>>>(x, y, t, sigma, g_w, g_b, out);
}